// LTAS_86320252715138
// MI455X (gfx1250) — compile-verified
//
#include <hip/hip_runtime.h>
#include <hip/hip_bf16.h>
#include <stdint.h>

#define D_MODEL   1024
#define NEG_SLOPE 0.2f

typedef __attribute__((ext_vector_type(2))) float v2f;
typedef __attribute__((ext_vector_type(8))) float v8f;

// ---- monotonic float<->uint mapping (ascending order preserved) ----
__device__ __forceinline__ unsigned map_asc(float f) {
  unsigned u = __float_as_uint(f);
  return (u & 0x80000000u) ? ~u : (u | 0x80000000u);
}
__device__ __forceinline__ float unmap_asc(unsigned u) {
  unsigned b = (u & 0x80000000u) ? (u & 0x7FFFFFFFu) : ~u;
  return __uint_as_float(b);
}

// ======================================================================
// 1) GEMV x = h @ W via V_WMMA_F32_16X16X4_F32.
//    One wave handles a 16-row tile; K walked in steps of 4.
//    A (16x4 f32, 2 VGPRs): lane L -> row = L&15; VGPR0/1 hold K = kb, kb+1
//    where kb = (L>>4)*2  (ISA 7.12.2, 32-bit A 16x4 layout).
//    B (4x16 f32, 2 VGPRs): lane L -> col = L&15; VGPR0/1 hold K = kb, kb+1.
//    Only col 0 is populated (out_channels == 1).
//    D (16x16 f32): column 0 lives in VGPR r of lane 0 (M=r) / lane 16 (M=8+r).
// ======================================================================
__global__ void gemv_wmma_kernel(const float* __restrict__ h,
                                 const float* __restrict__ W,
                                 float* __restrict__ x, int n) {
  const int wave = (int)((blockIdx.x * blockDim.x + threadIdx.x) >> 5);
  const int lane = (int)(threadIdx.x & 31u);
  const int row0 = wave * 16;
  if (row0 >= n) return;                       // wave-uniform: EXEC stays full

  const int m  = lane & 15;
  const int kb = (lane >> 4) << 1;             // 0 or 2
  int row = row0 + m;
  if (row >= n) row = n - 1;                   // clamp (ragged tail reads only)
  const float* __restrict__ arow = h + (size_t)row * D_MODEL;

  v8f c = {};
  for (int kk = 0; kk < D_MODEL; kk += 4) {
    v2f a, b;
    a.x = arow[kk + kb + 0];
    a.y = arow[kk + kb + 1];
    const float w0 = W[kk + kb + 0];
    const float w1 = W[kk + kb + 1];
    b.x = (m == 0) ? w0 : 0.0f;
    b.y = (m == 0) ? w1 : 0.0f;
    // (neg_a, A, neg_b, B, c_mod, C, reuse_a, reuse_b)
    c = __builtin_amdgcn_wmma_f32_16x16x4_f32(false, a, false, b,
                                              (short)0, c, false, false);
  }

  if (lane == 0) {
#pragma unroll
    for (int r = 0; r < 8; ++r) {
      const int rr = row0 + r;
      if (rr < n) x[rr] = c[r];
    }
  } else if (lane == 16) {
#pragma unroll
    for (int r = 0; r < 8; ++r) {
      const int rr = row0 + 8 + r;
      if (rr < n) x[rr] = c[r];
    }
  }
}

// ======================================================================
// 2) init per-node scratch
// ======================================================================
__global__ void init_kernel(unsigned* __restrict__ mkey,
                            float* __restrict__ denom,
                            float* __restrict__ numer, int n) {
  const int i = (int)(blockIdx.x * blockDim.x + threadIdx.x);
  if (i < n) {
    mkey[i]  = 0u;        // smallest possible mapped value (identity for max)
    denom[i] = 0.0f;
    numer[i] = 0.0f;
  }
}

// ======================================================================
// 3) edge pass 1: segment max of LeakyReLU logits (self-loops appended)
// ======================================================================
__global__ void edge_max_kernel(const int* __restrict__ src,
                                const int* __restrict__ dst,
                                const float* __restrict__ x,
                                const float* __restrict__ att_src,
                                const float* __restrict__ att_dst,
                                unsigned* __restrict__ mkey, int E, int n) {
  const int i = (int)(blockIdx.x * blockDim.x + threadIdx.x);
  if (i >= E + n) return;
  int s, d;
  if (i < E) {
    __builtin_prefetch(src + i + 4096, 0, 0);  // global_prefetch_b8
    s = src[i];
    d = dst[i];
  } else {
    s = d = i - E;
  }
  const float e0 = att_src[0] * x[s] + att_dst[0] * x[d];
  const float e  = (e0 >= 0.0f) ? e0 : NEG_SLOPE * e0;
  atomicMax(&mkey[d], map_asc(e));
}

// ======================================================================
// 4) edge pass 2: denom += exp(e-m),  numer += exp(e-m)*x[src]
// ======================================================================
__global__ void edge_sum_kernel(const int* __restrict__ src,
                                const int* __restrict__ dst,
                                const float* __restrict__ x,
                                const float* __restrict__ att_src,
                                const float* __restrict__ att_dst,
                                const unsigned* __restrict__ mkey,
                                float* __restrict__ denom,
                                float* __restrict__ numer, int E, int n) {
  const int i = (int)(blockIdx.x * blockDim.x + threadIdx.x);
  if (i >= E + n) return;
  int s, d;
  if (i < E) {
    __builtin_prefetch(dst + i + 4096, 0, 0);
    s = src[i];
    d = dst[i];
  } else {
    s = d = i - E;
  }
  const float e0 = att_src[0] * x[s] + att_dst[0] * x[d];
  const float e  = (e0 >= 0.0f) ? e0 : NEG_SLOPE * e0;
  const float ex = __expf(e - unmap_asc(mkey[d]));
  atomicAdd(&denom[d], ex);
  atomicAdd(&numer[d], ex * x[s]);
}

// ======================================================================
// 5) per-node epilogue: scores = numer/(denom+eps) + bias
// ======================================================================
__global__ void scores_kernel(const float* __restrict__ numer,
                              const float* __restrict__ denom,
                              const float* __restrict__ bias,
                              float* __restrict__ scores, int n) {
  const int i = (int)(blockIdx.x * blockDim.x + threadIdx.x);
  if (i < n) scores[i] = numer[i] / (denom[i] + 1e-16f) + bias[0];
}

// ======================================================================
// 6) build sort keys: descending-by-score, tie-break by index ascending
// ======================================================================
__global__ void build_keys_kernel(const float* __restrict__ scores,
                                  unsigned long long* __restrict__ keys,
                                  int n, int M) {
  const int i = (int)(blockIdx.x * blockDim.x + threadIdx.x);
  if (i >= M) return;
  if (i < n) {
    const unsigned k = ~map_asc(scores[i]);  // invert -> ascending sort == desc scores
    keys[i] = ((unsigned long long)k << 32) | (unsigned)i;
  } else {
    keys[i] = ~0ull;                         // padding sorts to the end
  }
}

// ======================================================================
// 7) bitonic compare-exchange step (ascending)
// ======================================================================
__global__ void bitonic_step_kernel(unsigned long long* __restrict__ a,
                                    int j, int k) {
  const int i   = (int)(blockIdx.x * blockDim.x + threadIdx.x);
  const int ixj = i ^ j;
  if (ixj > i) {
    const unsigned long long v0 = a[i];
    const unsigned long long v1 = a[ixj];
    const bool asc = ((i & k) == 0);
    if ((v0 > v1) == asc) {
      a[i]   = v1;
      a[ixj] = v0;
    }
  }
}

// ======================================================================
// 8) extract perm (int32 bit patterns into f32 output buffer)
// ======================================================================
__global__ void perm_kernel(const unsigned long long* __restrict__ keys,
                            int* __restrict__ perm, int n) {
  const int i = (int)(blockIdx.x * blockDim.x + threadIdx.x);
  if (i < n) perm[i] = (int)(keys[i] & 0xFFFFFFFFull);
}

// ======================================================================
// 9) gather: h_ordered[i,:] = h[perm[i],:]  (one row/block, float4/lane)
// ======================================================================
__global__ void gather_kernel(const float* __restrict__ h,
                              const unsigned long long* __restrict__ keys,
                              float* __restrict__ h_out, int n) {
  const int row = (int)blockIdx.x;
  int p = (int)(keys[row] & 0xFFFFFFFFull);
  if (p < 0 || p >= n) p = 0;  // defensive (padding never reaches row < n)
  const float4* __restrict__ s = (const float4*)(h + (size_t)p * D_MODEL);
  float4* __restrict__       d = (float4*)(h_out + (size_t)row * D_MODEL);
  d[threadIdx.x] = s[threadIdx.x];
}

// ======================================================================
extern "C" void kernel_launch(void* const* d_in, const int* in_sizes, int n_in,
                              void* d_out, int out_size, void* d_ws,
                              size_t ws_size, hipStream_t stream) {
  const float* h       = (const float*)d_in[0];
  const int*   ei      = (const int*)d_in[1];
  const float* W       = (const float*)d_in[2];
  const float* att_src = (const float*)d_in[3];
  const float* att_dst = (const float*)d_in[4];
  const float* bias    = (const float*)d_in[5];

  const int D = in_sizes[2];       // 1024
  const int n = in_sizes[0] / D;   // 100000
  const int E = in_sizes[1] / 2;   // 3,200,000
  const int* src = ei;
  const int* dst = ei + E;

  // ---- workspace carve-up (~2.7 MB) ----
  char* ws = (char*)d_ws;
  float*    x     = (float*)ws;    ws += (size_t)n * sizeof(float);
  unsigned* mkey  = (unsigned*)ws; ws += (size_t)n * sizeof(unsigned);
  float*    denom = (float*)ws;    ws += (size_t)n * sizeof(float);
  float*    numer = (float*)ws;    ws += (size_t)n * sizeof(float);
  uintptr_t pa = ((uintptr_t)ws + 7) & ~(uintptr_t)7;
  unsigned long long* keys = (unsigned long long*)pa;

  int M = 1;
  while (M < n) M <<= 1;           // 131072

  // ---- output carve-up: h_ordered [n*D] | perm [n] | scores [n] ----
  float* h_out  = (float*)d_out;
  int*   perm   = (int*)(h_out + (size_t)n * D);
  float* scores = (float*)(h_out + (size_t)n * D + n);

  // 1) GEMV (WMMA): one wave per 16-row tile
  {
    const int tiles   = (n + 15) / 16;
    const int threads = tiles * 32;
    gemv_wmma_kernel<<<(threads + 255) / 256, 256, 0, stream>>>(h, W, x, n);
  }
  // 2) init scratch
  init_kernel<<<(n + 255) / 256, 256, 0, stream>>>(mkey, denom, numer, n);
  // 3) segment max
  const int EN = E + n;
  edge_max_kernel<<<(EN + 255) / 256, 256, 0, stream>>>(src, dst, x, att_src,
                                                        att_dst, mkey, E, n);
  // 4) segment sums
  edge_sum_kernel<<<(EN + 255) / 256, 256, 0, stream>>>(
      src, dst, x, att_src, att_dst, mkey, denom, numer, E, n);
  // 5) scores
  scores_kernel<<<(n + 255) / 256, 256, 0, stream>>>(numer, denom, bias,
                                                     scores, n);
  // 6) sort keys
  build_keys_kernel<<<(M + 255) / 256, 256, 0, stream>>>(scores, keys, n, M);
  // 7) bitonic sort (L2-resident, 153 cheap stages)
  for (int k = 2; k <= M; k <<= 1)
    for (int j = k >> 1; j > 0; j >>= 1)
      bitonic_step_kernel<<<(M + 255) / 256, 256, 0, stream>>>(keys, j, k);
  // 8) perm
  perm_kernel<<<(n + 255) / 256, 256, 0, stream>>>(keys, perm, n);
  // 9) gather rows (dominant HBM cost: ~820 MB)
  gather_kernel<<<n, D / 4, 0, stream>>>(h, keys, h_out, n);
}